// RNNLM_90598040142179
// MI455X (gfx1250) — compile-verified
//
#include <hip/hip_runtime.h>

// RNNLM forward: out[s,b,v] = log_softmax( sigmoid(X@Wx + h0@Wh) @ Wo )
// V=32000, S=64, B=64, E=32, H=16. M = S*B = 4096 rows.
//
// Store-bound problem: output = 524 MB fp32 (~22.5us at 23.3 TB/s HBM).
// Wo (2MB) lives in L2 (192MB). Logits are never spilled to HBM: pass 1
// accumulates sum(exp(logit)) per row (|logit|<=4, so no max-shift needed),
// pass 2 recomputes the GEMM with V_WMMA_F32_16X16X4_F32 and writes
// logit - log(Z) once. K=16 fp32 GEMM = 4 chained f32 WMMAs per 16x16 tile.

#define VV 32000
#define SS 64
#define BB 64
#define EE 32
#define HH 16
#define MM (SS * BB)           // 4096 rows
#define NCB 25                 // column blocks over V
#define CB_COLS (VV / NCB)     // 1280 columns per block
#define WAVES 8                // 256-thread block = 8 wave32
#define STEP_COLS (WAVES * 16) // 128 columns per inner step
#define ITERS (CB_COLS / STEP_COLS) // 10

typedef __attribute__((ext_vector_type(2))) float v2f;
typedef __attribute__((ext_vector_type(8))) float v8f;

// ---------------------------------------------------------------------------
// Kernel 1: H_table[m][h] = sigmoid( lookup[tok[m]] . Wx[:,h] + (h0 . Wh[:,h]) )
// 65536 threads total; ~48 FMAs each. Negligible cost.
// ---------------------------------------------------------------------------
__global__ void hs_kernel(const int* __restrict__ tok,
                          const float* __restrict__ lookup,
                          const float* __restrict__ wx,
                          const float* __restrict__ wh,
                          const float* __restrict__ h0,
                          float* __restrict__ Hs) {
  int idx = blockIdx.x * blockDim.x + threadIdx.x; // [0, MM*HH)
  int m = idx >> 4;
  int h = idx & 15;
  float acc = 0.0f;
#pragma unroll
  for (int k = 0; k < HH; ++k) acc += h0[k] * wh[k * HH + h]; // h_rec (constant)
  const float* xr = lookup + (long)tok[m] * EE;
#pragma unroll
  for (int e = 0; e < EE; ++e) acc += xr[e] * wx[e * HH + h];
  Hs[m * HH + h] = 1.0f / (1.0f + __expf(-acc));
}

// ---------------------------------------------------------------------------
// WMMA tile helper: one wave computes a 16x16 fp32 logits tile for rows
// [m0, m0+16) x cols [col0, col0+16) via 4 chained V_WMMA_F32_16X16X4_F32.
// A frags (a[c]) are loaded once by the caller (invariant across col steps).
// B layout (ISA 7.12.2): VGPR0 = {K=0 lanes0-15, K=2 lanes16-31},
//                        VGPR1 = {K=1 lanes0-15, K=3 lanes16-31}.
// ---------------------------------------------------------------------------
__device__ __forceinline__ v8f logits_tile(const v2f a[4],
                                           const float* __restrict__ Wo,
                                           int col0, int ln, int half) {
  v8f acc = {0.f, 0.f, 0.f, 0.f, 0.f, 0.f, 0.f, 0.f};
  int col = col0 + ln;
#pragma unroll
  for (int c = 0; c < 4; ++c) {
    int k = 4 * c + 2 * half;
    v2f b;
    b.x = Wo[(long)k * VV + col];
    b.y = Wo[(long)(k + 1) * VV + col];
    acc = __builtin_amdgcn_wmma_f32_16x16x4_f32(
        /*neg_a=*/false, a[c], /*neg_b=*/false, b,
        /*c_mod=*/(short)0, acc, /*reuse_a=*/false, /*reuse_b=*/false);
  }
  return acc;
}

__device__ __forceinline__ void load_a_frags(v2f a[4],
                                             const float* __restrict__ Hs,
                                             int m0, int ln, int half) {
  // A 16x4 fp32 layout: lanes 0-15 hold K,K+1 for M=lane; lanes 16-31 hold K+2,K+3.
#pragma unroll
  for (int c = 0; c < 4; ++c) {
    int k = 4 * c + 2 * half;
    const float* hp = Hs + (m0 + ln) * HH + k;
    a[c].x = hp[0];
    a[c].y = hp[1];
  }
}

// ---------------------------------------------------------------------------
// Kernel 2: partials[row][cb] = sum_{v in colblock cb} exp(logits[row][v])
// Deterministic reduction: per-lane accumulators -> wave32 butterfly shuffle
// -> fixed-order LDS tree across the 8 waves (no float atomics).
// ---------------------------------------------------------------------------
__global__ void sumexp_kernel(const float* __restrict__ Hs,
                              const float* __restrict__ Wo,
                              float* __restrict__ partials) {
  const int tid = threadIdx.x;
  const int wave = tid >> 5;
  const int lane = tid & 31;
  const int half = lane >> 4;
  const int ln = lane & 15;
  const int m0 = blockIdx.y * 16;
  const int cb = blockIdx.x;

  v2f a[4];
  load_a_frags(a, Hs, m0, ln, half);

  float sacc[8] = {0.f, 0.f, 0.f, 0.f, 0.f, 0.f, 0.f, 0.f};
  for (int it = 0; it < ITERS; ++it) {
    int col0 = cb * CB_COLS + it * STEP_COLS + wave * 16;
    v8f acc = logits_tile(a, Wo, col0, ln, half);
#pragma unroll
    for (int r = 0; r < 8; ++r) sacc[r] += __expf(acc[r]); // |logit| <= 4: safe
  }

  // Reduce across the 16 lanes that share each row (xor masks stay in-half).
#pragma unroll
  for (int r = 0; r < 8; ++r) {
    float s = sacc[r];
#pragma unroll
    for (int msk = 1; msk < 16; msk <<= 1) s += __shfl_xor(s, msk, 32);
    sacc[r] = s;
  }

  __shared__ float wsum[WAVES][16];
  if (ln == 0) { // lane 0 (rows 0..7) and lane 16 (rows 8..15)
#pragma unroll
    for (int r = 0; r < 8; ++r) wsum[wave][half * 8 + r] = sacc[r];
  }
  __syncthreads();
  if (tid < 16) {
    float s = 0.f;
#pragma unroll
    for (int w = 0; w < WAVES; ++w) s += wsum[w][tid]; // fixed order: deterministic
    partials[(m0 + tid) * NCB + cb] = s;
  }
}

// ---------------------------------------------------------------------------
// Kernel 3: out[row][v] = logit[row][v] - log( sum_cb partials[row][cb] )
// Recomputes the WMMA tiles (cheap) so logits never round-trip through HBM.
// Stores are coalesced: per C-register, lanes 0-15 / 16-31 each write 16
// consecutive fp32 of one output row.
// ---------------------------------------------------------------------------
__global__ void out_kernel(const float* __restrict__ Hs,
                           const float* __restrict__ Wo,
                           const float* __restrict__ partials,
                           float* __restrict__ out) {
  const int tid = threadIdx.x;
  const int wave = tid >> 5;
  const int lane = tid & 31;
  const int half = lane >> 4;
  const int ln = lane & 15;
  const int m0 = blockIdx.y * 16;
  const int cb = blockIdx.x;

  v2f a[4];
  load_a_frags(a, Hs, m0, ln, half);

  // log(Z) for the 8 rows this lane's C registers map to (uniform control flow).
  float logZ[8];
#pragma unroll
  for (int r = 0; r < 8; ++r) {
    int row = m0 + r + 8 * half;
    float z = 0.f;
#pragma unroll
    for (int c = 0; c < NCB; ++c) z += partials[row * NCB + c];
    logZ[r] = __logf(z);
  }

  for (int it = 0; it < ITERS; ++it) {
    int col0 = cb * CB_COLS + it * STEP_COLS + wave * 16;
    v8f acc = logits_tile(a, Wo, col0, ln, half);
    int col = col0 + ln;
#pragma unroll
    for (int r = 0; r < 8; ++r) {
      int row = m0 + r + 8 * half;
      out[(long)row * VV + col] = acc[r] - logZ[r];
    }
  }
}

// ---------------------------------------------------------------------------
extern "C" void kernel_launch(void* const* d_in, const int* in_sizes, int n_in,
                              void* d_out, int out_size, void* d_ws, size_t ws_size,
                              hipStream_t stream) {
  const int*   tok    = (const int*)d_in[0];   // [S,B] int32
  const float* lookup = (const float*)d_in[1]; // [V,E]
  const float* wx     = (const float*)d_in[2]; // [E,H]
  const float* wh     = (const float*)d_in[3]; // [H,H]
  const float* wo     = (const float*)d_in[4]; // [H,V]
  const float* h0     = (const float*)d_in[5]; // [H]
  float* out = (float*)d_out;                  // [S,B,V]

  // Workspace: Hs = 4096*16 fp32 (256KB), partials = 4096*25 fp32 (400KB).
  float* Hs = (float*)d_ws;
  float* partials = Hs + MM * HH;

  hs_kernel<<<(MM * HH) / 256, 256, 0, stream>>>(tok, lookup, wx, wh, h0, Hs);
  sumexp_kernel<<<dim3(NCB, MM / 16), 256, 0, stream>>>(Hs, wo, partials);
  out_kernel<<<dim3(NCB, MM / 16), 256, 0, stream>>>(Hs, wo, partials, out);
}